// VDBEstimator_1726576856130
// MI455X (gfx1250) — compile-verified
//
#include <hip/hip_runtime.h>
#include <stdint.h>

#define ITEMS 8
#define BLOCK 256
#define TILE (BLOCK * ITEMS)   // 2048 elements per block
#define NWAVES (BLOCK / 32)
#define EPS 1e-4f

typedef unsigned int v4u __attribute__((ext_vector_type(4)));
typedef int          v8i __attribute__((ext_vector_type(8)));
typedef int          v4i __attribute__((ext_vector_type(4)));

struct Pair { float v; int f; };

__device__ __forceinline__ Pair comb(Pair a, Pair b) {
    Pair r;
    r.v = b.f ? b.v : a.v * b.v;
    r.f = a.f | b.f;
    return r;
}

// ---------------------------------------------------------------------------
// Tensor Data Mover: DMA a 1-D tile of 4-byte elements from global into LDS.
// D# layout per CDNA5 ISA (group0: count/lds/global/type,
// group1: data_size, tensor_dim0, tile dims, strides).
// This toolchain exposes the 6-arg builtin:
//   (uint32x4 g0, int32x8 g1, int32x4, int32x4, int32x8, i32 cpol)
// ---------------------------------------------------------------------------
__device__ __forceinline__ void tdm_load_tile_b32(uint32_t lds_byte_off,
                                                  const void* gptr,
                                                  uint32_t tile_elems,
                                                  uint32_t tensor_elems) {
    uint64_t ga = (uint64_t)(uintptr_t)gptr;
    v4u g0;
    g0.x = 1u;                                              // count=1 (valid user D#)
    g0.y = lds_byte_off;                                    // lds_addr
    g0.z = (uint32_t)ga;                                    // global_addr[31:0]
    g0.w = (uint32_t)((ga >> 32) & 0x1FFFFFFu) | (2u << 30);// global_addr[56:32] | type=2

    v8i g1;
    uint32_t td0 = tensor_elems;    // tensor_dim0 (data_size units), OOB reads -> 0
    uint32_t tl0 = tile_elems;      // tile_dim0
    g1[0] = (int)(2u << 16);                                  // data_size = 4B (code 2)
    g1[1] = (int)((td0 & 0xFFFFu) << 16);                     // tensor_dim0[15:0]
    g1[2] = (int)(((td0 >> 16) & 0xFFFFu) | (1u << 16));      // tensor_dim0[31:16] | tensor_dim1=1
    g1[3] = (int)((tl0 & 0xFFFFu) << 16);                     // tensor_dim1 hi=0 | tile_dim0
    g1[4] = (int)1;                                           // tile_dim1=1, tile_dim2=0
    g1[5] = (int)td0;                                         // tensor_dim0_stride[31:0]
    g1[6] = 0;                                                // stride0 hi / stride1 lo
    g1[7] = 0;                                                // stride1 hi

    v4i z4 = {0, 0, 0, 0};
    v8i z8 = {0, 0, 0, 0, 0, 0, 0, 0};
    __builtin_amdgcn_tensor_load_to_lds(g0, g1, z4, z4, z8, 0);
}

// Wave32 Hillis-Steele inclusive segmented scan (non-commutative op is fine:
// each step folds the left-neighbor partial on the left).
__device__ __forceinline__ Pair wave_incl_scan(Pair p, int lane) {
    #pragma unroll
    for (int d = 1; d < 32; d <<= 1) {
        float pv = __shfl_up(p.v, d);
        int   pf = __shfl_up(p.f, d);
        if (lane >= d) { Pair q; q.v = pv; q.f = pf; p = comb(q, p); }
    }
    return p;
}

// Build element (y, first) for global index g (local l within tile), updating
// the running previous alpha/idx.
__device__ __forceinline__ Pair make_elem(const float* s_a, const int* s_i,
                                          int g, int l, int n,
                                          float& aprev, int& iprev) {
    Pair e;
    if (g < n) {
        int cur   = s_i[l];
        int first = (g == 0) | (cur != iprev);
        e.v = first ? 1.0f : (1.0f - aprev);
        e.f = first;
        aprev = s_a[l];
        iprev = cur;
    } else {
        e.v = 1.0f; e.f = 1;     // padding: neutral segment start, never stored
    }
    return e;
}

// ---------------------------------------------------------------------------
// Kernel 1: per-block segmented-scan aggregate (v,f) -> workspace
// ---------------------------------------------------------------------------
__global__ void k_block_agg(const float* __restrict__ alphas,
                            const int* __restrict__ ridx,
                            int n,
                            float* __restrict__ agg_v,
                            int* __restrict__ agg_f) {
    __shared__ __align__(16) unsigned char smem[TILE * 4 + TILE * 4 + NWAVES * 8];
    float* s_a  = (float*)smem;
    int*   s_i  = (int*)(smem + TILE * 4);
    float* s_wv = (float*)(smem + TILE * 8);
    int*   s_wf = (int*)(smem + TILE * 8 + NWAVES * 4);

    const int b    = blockIdx.x;
    const int base = b * TILE;
    const int tid  = threadIdx.x;
    const int lane = tid & 31;
    const int wave = tid >> 5;

    if (tid < 32) {   // wave 0 drives the DMA (TDM ignores EXEC; issue per wave)
        uint32_t rem = (uint32_t)(n - base);
        tdm_load_tile_b32(0,        alphas + base, TILE, rem);
        tdm_load_tile_b32(TILE * 4, ridx   + base, TILE, rem);
        __builtin_amdgcn_s_wait_tensorcnt(0);
    }
    __syncthreads();

    const int l0 = tid * ITEMS;
    float aprev; int iprev;
    if (l0 == 0) {
        if (base == 0) { aprev = 0.0f; iprev = -1; }
        else           { aprev = alphas[base - 1]; iprev = ridx[base - 1]; }
    } else {
        aprev = s_a[l0 - 1]; iprev = s_i[l0 - 1];
    }

    Pair acc; acc.v = 1.0f; acc.f = 0;
    #pragma unroll
    for (int k = 0; k < ITEMS; ++k)
        acc = comb(acc, make_elem(s_a, s_i, base + l0 + k, l0 + k, n, aprev, iprev));

    Pair ws = wave_incl_scan(acc, lane);
    if (lane == 31) { s_wv[wave] = ws.v; s_wf[wave] = ws.f; }
    __syncthreads();

    if (tid == 0) {
        Pair t; t.v = 1.0f; t.f = 0;
        #pragma unroll
        for (int w = 0; w < NWAVES; ++w) { Pair q; q.v = s_wv[w]; q.f = s_wf[w]; t = comb(t, q); }
        agg_v[b] = t.v; agg_f[b] = t.f;
    }
}

// ---------------------------------------------------------------------------
// Kernel 2: single-block exclusive segmented scan over the block aggregates
// ---------------------------------------------------------------------------
__global__ void k_scan_aggs(const float* __restrict__ agg_v,
                            const int* __restrict__ agg_f,
                            float* __restrict__ car_v,
                            int* __restrict__ car_f,
                            int nblocks) {
    __shared__ float s_wv[NWAVES]; __shared__ int s_wf[NWAVES];
    __shared__ float s_ev[NWAVES]; __shared__ int s_ef[NWAVES];

    const int tid  = threadIdx.x;
    const int lane = tid & 31;
    const int wave = tid >> 5;
    const int chunk = (nblocks + BLOCK - 1) / BLOCK;
    const int st = tid * chunk;
    const int en = min(st + chunk, nblocks);

    Pair acc; acc.v = 1.0f; acc.f = 0;
    for (int i = st; i < en; ++i) { Pair q; q.v = agg_v[i]; q.f = agg_f[i]; acc = comb(acc, q); }

    Pair incl = wave_incl_scan(acc, lane);
    float pv = __shfl_up(incl.v, 1);
    int   pf = __shfl_up(incl.f, 1);
    Pair texc; if (lane == 0) { texc.v = 1.0f; texc.f = 0; } else { texc.v = pv; texc.f = pf; }

    if (lane == 31) { s_wv[wave] = incl.v; s_wf[wave] = incl.f; }
    __syncthreads();
    if (tid == 0) {
        Pair run; run.v = 1.0f; run.f = 0;
        #pragma unroll
        for (int w = 0; w < NWAVES; ++w) {
            s_ev[w] = run.v; s_ef[w] = run.f;
            Pair q; q.v = s_wv[w]; q.f = s_wf[w];
            run = comb(run, q);
        }
    }
    __syncthreads();

    Pair wex; wex.v = s_ev[wave]; wex.f = s_ef[wave];
    Pair ex = comb(wex, texc);
    for (int i = st; i < en; ++i) {
        car_v[i] = ex.v; car_f[i] = ex.f;
        Pair q; q.v = agg_v[i]; q.f = agg_f[i];
        ex = comb(ex, q);
    }
}

// ---------------------------------------------------------------------------
// Kernel 3: recompute local scan seeded with combine(carry, blockExclusive),
// write trans and vis.
// ---------------------------------------------------------------------------
__global__ void k_apply(const float* __restrict__ alphas,
                        const int* __restrict__ ridx,
                        int n,
                        const float* __restrict__ car_v,
                        const int* __restrict__ car_f,
                        float* __restrict__ trans,
                        float* __restrict__ vis) {
    __shared__ __align__(16) unsigned char smem[TILE * 4 + TILE * 4 + 4 * NWAVES * 4];
    float* s_a  = (float*)smem;
    int*   s_i  = (int*)(smem + TILE * 4);
    float* s_wv = (float*)(smem + TILE * 8);
    int*   s_wf = (int*)(smem + TILE * 8 + NWAVES * 4);
    float* s_ev = (float*)(smem + TILE * 8 + NWAVES * 8);
    int*   s_ef = (int*)(smem + TILE * 8 + NWAVES * 12);

    const int b    = blockIdx.x;
    const int base = b * TILE;
    const int tid  = threadIdx.x;
    const int lane = tid & 31;
    const int wave = tid >> 5;

    if (tid < 32) {
        uint32_t rem = (uint32_t)(n - base);
        tdm_load_tile_b32(0,        alphas + base, TILE, rem);
        tdm_load_tile_b32(TILE * 4, ridx   + base, TILE, rem);
        __builtin_amdgcn_s_wait_tensorcnt(0);
    }
    __syncthreads();

    const int l0 = tid * ITEMS;
    float aprev0; int iprev0;
    if (l0 == 0) {
        if (base == 0) { aprev0 = 0.0f; iprev0 = -1; }
        else           { aprev0 = alphas[base - 1]; iprev0 = ridx[base - 1]; }
    } else {
        aprev0 = s_a[l0 - 1]; iprev0 = s_i[l0 - 1];
    }

    // Pass 1: thread aggregate
    float aprev = aprev0; int iprev = iprev0;
    Pair acc; acc.v = 1.0f; acc.f = 0;
    #pragma unroll
    for (int k = 0; k < ITEMS; ++k)
        acc = comb(acc, make_elem(s_a, s_i, base + l0 + k, l0 + k, n, aprev, iprev));

    // Scan hierarchy -> per-thread exclusive prefix within the block
    Pair incl = wave_incl_scan(acc, lane);
    float pv = __shfl_up(incl.v, 1);
    int   pf = __shfl_up(incl.f, 1);
    Pair texc; if (lane == 0) { texc.v = 1.0f; texc.f = 0; } else { texc.v = pv; texc.f = pf; }

    if (lane == 31) { s_wv[wave] = incl.v; s_wf[wave] = incl.f; }
    __syncthreads();
    if (tid == 0) {
        Pair run; run.v = 1.0f; run.f = 0;
        #pragma unroll
        for (int w = 0; w < NWAVES; ++w) {
            s_ev[w] = run.v; s_ef[w] = run.f;
            Pair q; q.v = s_wv[w]; q.f = s_wf[w];
            run = comb(run, q);
        }
    }
    __syncthreads();

    Pair wex;   wex.v = s_ev[wave]; wex.f = s_ef[wave];
    Pair carry; carry.v = car_v[b]; carry.f = car_f[b];
    Pair C = comb(carry, comb(wex, texc));

    // Pass 2: fold C into the thread-serial scan and emit results
    aprev = aprev0; iprev = iprev0;
    float tbuf[ITEMS], vbuf[ITEMS];
    Pair run = C;
    #pragma unroll
    for (int k = 0; k < ITEMS; ++k) {
        run = comb(run, make_elem(s_a, s_i, base + l0 + k, l0 + k, n, aprev, iprev));
        tbuf[k] = run.v;
        vbuf[k] = (run.v >= EPS) ? 1.0f : 0.0f;
    }

    const int g0 = base + l0;
    if (g0 + ITEMS <= n) {   // full vector path: 2x float4 per output array
        float4* tp = (float4*)(trans + g0);
        float4* vp = (float4*)(vis + g0);
        tp[0] = make_float4(tbuf[0], tbuf[1], tbuf[2], tbuf[3]);
        tp[1] = make_float4(tbuf[4], tbuf[5], tbuf[6], tbuf[7]);
        vp[0] = make_float4(vbuf[0], vbuf[1], vbuf[2], vbuf[3]);
        vp[1] = make_float4(vbuf[4], vbuf[5], vbuf[6], vbuf[7]);
    } else {
        for (int k = 0; k < ITEMS; ++k) {
            int g = g0 + k;
            if (g < n) { trans[g] = tbuf[k]; vis[g] = vbuf[k]; }
        }
    }
}

// ---------------------------------------------------------------------------
extern "C" void kernel_launch(void* const* d_in, const int* in_sizes, int n_in,
                              void* d_out, int out_size, void* d_ws, size_t ws_size,
                              hipStream_t stream) {
    const float* alphas = (const float*)d_in[0];
    const int*   ridx   = (const int*)d_in[1];   // integer input -> const int* per harness convention
    (void)n_in; (void)out_size; (void)ws_size;

    const int n       = in_sizes[0];
    const int nblocks = (n + TILE - 1) / TILE;   // 4096 for N = 8,388,608

    char* ws = (char*)d_ws;
    float* agg_v = (float*)(ws);
    int*   agg_f = (int*)(ws + (size_t)nblocks * 4);
    float* car_v = (float*)(ws + (size_t)nblocks * 8);
    int*   car_f = (int*)(ws + (size_t)nblocks * 12);

    float* trans = (float*)d_out;
    float* vis   = trans + n;

    k_block_agg<<<nblocks, BLOCK, 0, stream>>>(alphas, ridx, n, agg_v, agg_f);
    k_scan_aggs<<<1, BLOCK, 0, stream>>>(agg_v, agg_f, car_v, car_f, nblocks);
    k_apply<<<nblocks, BLOCK, 0, stream>>>(alphas, ridx, n, car_v, car_f, trans, vis);
}